// SoMLoss_25305947308623
// MI455X (gfx1250) — compile-verified
//
#include <hip/hip_runtime.h>

typedef __attribute__((ext_vector_type(2))) float v2f;
typedef __attribute__((ext_vector_type(8))) float v8f;

#define MARGIN_F 1.0f
#define RANKING_W 1.0f
#define CONTRASTIVE_W 0.5f
#define REV_GEOM_W 0.3f
#define ITPB 4  // i-tiles (16 rows each) per block

__device__ __forceinline__ float fix_inf(float v) {
    return __builtin_isinf(v) ? -100.0f : v;
}

// relu in one VALU op: v_med3_f32(x, 0, +inf) == clamp(x, 0, inf)
__device__ __forceinline__ float relu1(float v) {
    return __builtin_amdgcn_fmed3f(v, 0.0f, __builtin_inff());
}

// ---------------- workspace accumulators ----------------
// acc[0] = ranking weighted pair sum
// acc[1] = ranking pair count  (sum_b som_b * nonsom_b, exact < 2^24)
// acc[2] = sum over batches of gated contrastive loss
// acc[3] = sum of bce * valid
// acc[4] = sum of valid (BCE denominator)

__global__ void som_init_kernel(float* acc) {
    for (int i = 0; i < 8; ++i) acc[i] = 0.0f;
}

__device__ float block_reduce_sum(float v, float* red) {
    const int tid = threadIdx.x;
    red[tid] = v;
    __syncthreads();
    for (int s = blockDim.x >> 1; s > 0; s >>= 1) {
        if (tid < s) red[tid] += red[tid + s];
        __syncthreads();
    }
    float r = red[0];
    __syncthreads();
    return r;
}

__device__ float block_reduce_max(float v, float* red) {
    const int tid = threadIdx.x;
    red[tid] = v;
    __syncthreads();
    for (int s = blockDim.x >> 1; s > 0; s >>= 1) {
        if (tid < s) red[tid] = fmaxf(red[tid], red[tid + s]);
        __syncthreads();
    }
    float r = red[0];
    __syncthreads();
    return r;
}

// ------------- per-batch contrastive + BCE stats + pair count (O(B*N)) -------------
__global__ void som_stats_kernel(const float* __restrict__ scores,
                                 const float* __restrict__ pose,
                                 const float* __restrict__ som,
                                 const unsigned char* __restrict__ valid,
                                 float* __restrict__ acc, int N) {
    __shared__ float red[256];
    const int b = blockIdx.x;
    const int tid = threadIdx.x;
    const int T = blockDim.x;
    const size_t base = (size_t)b * (size_t)N;

    // pass 1: row max of masked logits, target sums, counts, BCE
    float lmax = -3.0e38f;
    float tsum = 0.f, somc = 0.f, nonsomc = 0.f, vc = 0.f, bsum = 0.f;
    for (int j = tid; j < N; j += T) {
        float s = fix_inf(scores[base + j]);
        float y = som[base + j];
        float vm = valid[base + j] ? 1.f : 0.f;
        float logit = (vm > 0.f) ? s : -1.0e9f;
        lmax = fmaxf(lmax, logit);
        tsum += y * vm;
        somc += (y > 0.f) ? vm : 0.f;
        nonsomc += (y == 0.f) ? vm : 0.f;
        vc += vm;
        float x = pose[base + j];
        float bce = fmaxf(x, 0.f) - x * y + log1pf(expf(-fabsf(x)));
        bsum += bce * vm;
    }
    lmax    = block_reduce_max(lmax, red);
    tsum    = block_reduce_sum(tsum, red);
    somc    = block_reduce_sum(somc, red);
    nonsomc = block_reduce_sum(nonsomc, red);
    vc      = block_reduce_sum(vc,   red);
    bsum    = block_reduce_sum(bsum, red);

    // pass 2: softmax denominator
    float zpart = 0.f;
    for (int j = tid; j < N; j += T) {
        float s = fix_inf(scores[base + j]);
        float vm = valid[base + j] ? 1.f : 0.f;
        float logit = (vm > 0.f) ? s : -1.0e9f;
        zpart += expf(logit - lmax);
    }
    float Z = block_reduce_sum(zpart, red);

    // pass 3: -(target_dist * log(probs + 1e-8))
    float tden = fmaxf(tsum, 1e-12f);
    float lb = 0.f;
    for (int j = tid; j < N; j += T) {
        float s = fix_inf(scores[base + j]);
        float y = som[base + j];
        float vm = valid[base + j] ? 1.f : 0.f;
        float logit = (vm > 0.f) ? s : -1.0e9f;
        float p = expf(logit - lmax) / Z;
        float td = (y * vm) / tden;
        lb += -td * logf(p + 1e-8f);
    }
    lb = block_reduce_sum(lb, red);

    if (tid == 0) {
        bool gate = (somc > 0.f) && (vc > 1.f) && (tsum > 0.f);
        if (gate) atomicAdd(&acc[2], lb);
        atomicAdd(&acc[1], somc * nonsomc);  // separable pair count
        atomicAdd(&acc[3], bsum);
        atomicAdd(&acc[4], vc);
    }
}

// ------------- ranking loss via WMMA pair tiles (O(B*N^2)) -------------
// Masks folded into operands (rm,cm in {0,1} => relu(rm*cm*x) = rm*cm*relu(x)):
//   A[i] = [rm_i*(margin - s_i), rm_i, 0, 0]
//   B[:,j] = [cm_j, cm_j*s_j, 0, 0]
//   D[i][j] = rm_i*cm_j*(margin - s_i + s_j)
// One v_wmma_f32_16x16x4_f32 builds 256 masked pair values; epilogue is a
// single v_med3 (relu) + packed add per element. 4 i-tiles per block share
// each column load; j-loop has a uniform trip count so EXEC stays all-ones.
__global__ void som_rank_kernel(const float* __restrict__ scores,
                                const float* __restrict__ som,
                                const unsigned char* __restrict__ valid,
                                float* __restrict__ acc, int N) {
    const int b = blockIdx.x;
    const int it0 = blockIdx.y * ITPB;
    const size_t base = (size_t)b * (size_t)N;

    __shared__ float sh_a0[ITPB * 16];  // rm * (margin - s_i)
    __shared__ float sh_a1[ITPB * 16];  // rm

    const int tid = threadIdx.x;
    if (tid < ITPB * 16) {
        int row = it0 * 16 + tid;
        float s = fix_inf(scores[base + row]);
        float y = som[base + row];
        float vm = valid[base + row] ? 1.f : 0.f;
        float rm = (y > 0.f) ? vm : 0.f;  // som_mask row
        sh_a0[tid] = rm * (MARGIN_F - s);
        sh_a1[tid] = rm;
    }
    __syncthreads();

    const int lane = tid & 31;
    const int wave = tid >> 5;
    const int nwaves = blockDim.x >> 5;
    const bool lo = (lane < 16);
    const int l15 = lane & 15;

    // A operands: lanes 0-15 hold K=0/K=1 in VGPR0/1, lanes 16-31 K=2/K=3 (zero pad)
    v2f a[ITPB];
#pragma unroll
    for (int t = 0; t < ITPB; ++t) {
        a[t].x = lo ? sh_a0[t * 16 + l15] : 0.f;
        a[t].y = lo ? sh_a1[t * 16 + l15] : 0.f;
    }

    const v8f czero = {0.f, 0.f, 0.f, 0.f, 0.f, 0.f, 0.f, 0.f};
    float accV[8];
#pragma unroll
    for (int k = 0; k < 8; ++k) accV[k] = 0.f;

    const int njt = N >> 4;            // 128 j-tiles
    const int iters = njt / nwaves;    // exact: 128 % nwaves == 0
    for (int it = 0; it < iters; ++it) {
        const int jt = wave + it * nwaves;
        const int col = jt * 16 + l15;
        float sj = fix_inf(scores[base + col]);
        float yj = som[base + col];
        float vj = valid[base + col] ? 1.f : 0.f;
        float cm = (yj == 0.f) ? vj : 0.f;  // non_som_mask col

        v2f bm;
        bm.x = lo ? cm : 0.f;        // K=0 row of B = cm
        bm.y = lo ? cm * sj : 0.f;   // K=1 row of B = cm * s_j

#pragma unroll
        for (int t = 0; t < ITPB; ++t) {
            v8f d = __builtin_amdgcn_wmma_f32_16x16x4_f32(
                false, a[t], false, bm, (short)0, czero, false, false);
#pragma unroll
            for (int k = 0; k < 8; ++k) {
                accV[k] += relu1(d[k]);  // masked relu already in d
            }
        }
    }

    float accS = ((accV[0] + accV[1]) + (accV[2] + accV[3])) +
                 ((accV[4] + accV[5]) + (accV[6] + accV[7]));
#pragma unroll
    for (int o = 16; o > 0; o >>= 1) {
        accS += __shfl_xor(accS, o, 32);
    }
    if (lane == 0) {
        atomicAdd(&acc[0], accS);
    }
}

// ------------- finalize -------------
__global__ void som_finalize_kernel(const float* __restrict__ acc,
                                    float* __restrict__ out, int B) {
    float cnt = acc[1];
    float ranking = (cnt > 0.f) ? (acc[0] / fmaxf(cnt, 1.f)) : 0.f;
    float contrastive = acc[2] / (float)B;
    float pose_loss = acc[3] / fmaxf(acc[4], 1.f);
    out[0] = RANKING_W * ranking + CONTRASTIVE_W * contrastive + REV_GEOM_W * pose_loss;
}

extern "C" void kernel_launch(void* const* d_in, const int* in_sizes, int n_in,
                              void* d_out, int out_size, void* d_ws, size_t ws_size,
                              hipStream_t stream) {
    const float* scores = (const float*)d_in[0];
    const float* pose   = (const float*)d_in[1];
    // d_in[2] (chemistry_scores) is unused by the reference
    const float* som    = (const float*)d_in[3];
    const unsigned char* valid = (const unsigned char*)d_in[4];

    const int N = 2048;
    const int total = in_sizes[0];
    const int B = total / N;

    float* acc = (float*)d_ws;

    som_init_kernel<<<1, 1, 0, stream>>>(acc);
    som_stats_kernel<<<B, 256, 0, stream>>>(scores, pose, som, valid, acc, N);
    dim3 grid(B, N / (16 * ITPB));
    som_rank_kernel<<<grid, 256, 0, stream>>>(scores, som, valid, acc, N);
    som_finalize_kernel<<<1, 1, 0, stream>>>(acc, (float*)d_out, B);
}